// _NonLocalBlockND_23304492548806
// MI455X (gfx1250) — compile-verified
//
#include <hip/hip_runtime.h>
#include <hip/hip_bf16.h>

typedef float v2f __attribute__((ext_vector_type(2)));
typedef float v8f __attribute__((ext_vector_type(8)));

#define WMMA_F32X4(a, b, c) \
    __builtin_amdgcn_wmma_f32_16x16x4_f32(false, (a), false, (b), (short)0, (c), false, false)

constexpr int B  = 8;
constexpr int C  = 256;
constexpr int IC = 128;
constexpr int H  = 64;
constexpr int W  = 64;
constexpr int N  = H * W;            // 4096
constexpr int M  = (H / 2) * (W / 2); // 1024
constexpr float BN_EPS = 1e-5f;

// workspace layout in floats (~25.3 MB total)
constexpr size_t GP_OFF   = 0;                              // g_pool   B*IC*M
constexpr size_t PP_OFF   = GP_OFF + (size_t)B * IC * M;    // phi_pool B*IC*M
constexpr size_t ST_OFF   = PP_OFF + (size_t)B * IC * M;    // s_t      B*N
constexpr size_t SP_OFF   = ST_OFF + (size_t)B * N;         // s_p      B*M
constexpr size_t TW_OFF   = SP_OFF + (size_t)B * M;         // tw_eff   C
constexpr size_t TB_OFF   = TW_OFF + C;                     // tb_eff   (padded)
constexpr size_t MEAN_OFF = TB_OFF + 256;                   // mean     C
constexpr size_t RSTD_OFF = MEAN_OFF + C;                   // rstd     C
constexpr size_t Y_OFF    = RSTD_OFF + C + 256;             // Y        B*IC*N

// ---------------------------------------------------------------------------
// Collapse theta branch: tw_eff[c] = sum_ic cp_w[ic]*theta_w[ic,c]; tb_eff.
__global__ void k_redw(const float* __restrict__ cp_w, const float* __restrict__ th_w,
                       const float* __restrict__ th_b, float* __restrict__ tw,
                       float* __restrict__ tb) {
    int c = threadIdx.x;  // 256 threads
    float acc = 0.f;
    for (int ic = 0; ic < IC; ++ic) acc += cp_w[ic] * th_w[ic * C + c];
    tw[c] = acc;
    if (c == 0) {
        float a = 0.f;
        for (int ic = 0; ic < IC; ++ic) a += cp_w[ic] * th_b[ic];
        *tb = a;
    }
}

// s_t[b,n] = tb_eff + sum_c tw_eff[c] * x[b,c,n]
__global__ void k_st(const float* __restrict__ x, const float* __restrict__ tw,
                     const float* __restrict__ tb, float* __restrict__ st) {
    int t = blockIdx.x * blockDim.x + threadIdx.x;  // B*N threads
    int b = t >> 12;
    int n = t & (N - 1);
    float acc = *tb;
    const float* xp = x + (size_t)b * C * N + n;
    for (int c = 0; c < C; ++c) acc += tw[c] * xp[(size_t)c * N];
    st[t] = acc;
}

// ---------------------------------------------------------------------------
// g & phi 1x1 conv (WMMA f32 16x16x4) with fused 2x2 maxpool.
// One wave -> one 16(ic) x 16(pooled-m) tile = four full-res 16x16 tiles.
__global__ void __launch_bounds__(256) k_gphi(
    const float* __restrict__ x, const float* __restrict__ g_w,
    const float* __restrict__ g_b, const float* __restrict__ ph_w,
    const float* __restrict__ ph_b, float* __restrict__ gp, float* __restrict__ pp) {
    int wave = (blockIdx.x * blockDim.x + threadIdx.x) >> 5;  // 0..4095
    int lane = threadIdx.x & 31;
    int lo = lane & 15, hi = lane >> 4;

    int b    = wave >> 9;         // 512 waves per batch (8 o-tiles * 64 m-tiles)
    int ot   = (wave >> 6) & 7;   // which 16-row slab of IC
    int mt   = wave & 63;         // pooled-column tile
    int mh   = mt >> 1;           // pooled row 0..31
    int half = mt & 1;            // which 16 pooled cols of the 32-wide row
    int mw0  = half * 16;
    int h0 = 2 * mh, w0 = 2 * mw0;
    int orow = ot * 16;

    const float* xb = x + (size_t)b * C * N;

    v8f ag[2][2], ap[2][2];
#pragma unroll
    for (int r = 0; r < 2; ++r)
#pragma unroll
        for (int t = 0; t < 2; ++t)
#pragma unroll
            for (int v = 0; v < 8; ++v) {
                ag[r][t][v] = g_b[orow + 8 * hi + v];
                ap[r][t][v] = ph_b[orow + 8 * hi + v];
            }

    int nbase = h0 * W + w0 + lo;  // full-res flat index base for this lane

    for (int k = 0; k < C; k += 4) {
        int kk = k + 2 * hi;  // this lane's K pair within the 4-chunk
        v2f a_g = *(const v2f*)&g_w[(orow + lo) * C + kk];
        v2f a_p = *(const v2f*)&ph_w[(orow + lo) * C + kk];
#pragma unroll
        for (int r = 0; r < 2; ++r)
#pragma unroll
            for (int t = 0; t < 2; ++t) {
                int n = nbase + r * W + 16 * t;
                v2f bf;
                bf.x = xb[(size_t)kk * N + n];
                bf.y = xb[(size_t)(kk + 1) * N + n];
                ag[r][t] = WMMA_F32X4(a_g, bf, ag[r][t]);
                ap[r][t] = WMMA_F32X4(a_p, bf, ap[r][t]);
            }
    }

    // 2x2 maxpool: rows via v_max across the two h fragments,
    // columns via lane-xor-1 shuffle (adjacent w in adjacent lanes).
    bool writer = ((lo & 1) == 0);
#pragma unroll
    for (int t = 0; t < 2; ++t) {
        int m = mh * 32 + mw0 + 8 * t + (lo >> 1);
#pragma unroll
        for (int v = 0; v < 8; ++v) {
            float gm = fmaxf(ag[0][t][v], ag[1][t][v]);
            gm = fmaxf(gm, __shfl_xor(gm, 1, 32));
            float pm = fmaxf(ap[0][t][v], ap[1][t][v]);
            pm = fmaxf(pm, __shfl_xor(pm, 1, 32));
            if (writer) {
                size_t row = (size_t)(b * IC + orow + 8 * hi + v);
                gp[row * M + m] = gm;
                pp[row * M + m] = pm;
            }
        }
    }
}

// s_p[b,m] = sum_ic cp_w[IC+ic] * phi_pool[b,ic,m]
__global__ void k_sp(const float* __restrict__ pp, const float* __restrict__ cp_w,
                     float* __restrict__ sp) {
    int t = blockIdx.x * blockDim.x + threadIdx.x;  // B*M threads
    int b = t >> 10;
    int m = t & (M - 1);
    const float* wp = cp_w + IC;
    const float* p = pp + (size_t)b * IC * M + m;
    float acc = 0.f;
    for (int ic = 0; ic < IC; ++ic) acc += wp[ic] * p[(size_t)ic * M];
    sp[t] = acc;
}

// ---------------------------------------------------------------------------
// y[b,n,c] = (1/M) sum_m relu(s_t[n]+s_p[m]) * g_pool[b,c,m]  (WMMA f32)
// Block = one (b, n-tile); its 8 waves cover the 8 c-tiles; s_p staged in LDS.
// Y stored as (B, IC, N) so the next conv reads it as a B-matrix directly.
__global__ void __launch_bounds__(256) k_attn(const float* __restrict__ gp,
                                              const float* __restrict__ st,
                                              const float* __restrict__ sp,
                                              float* __restrict__ Y) {
    __shared__ float sps[M];
    int b  = blockIdx.x >> 8;
    int nt = blockIdx.x & 255;
    for (int i = threadIdx.x; i < M; i += 256) sps[i] = sp[b * M + i];
    __syncthreads();

    int ct   = threadIdx.x >> 5;  // c-tile 0..7
    int lane = threadIdx.x & 31;
    int lo = lane & 15, hi = lane >> 4;

    float stl = st[b * N + nt * 16 + lo];
    const float invM = 1.0f / (float)M;
    const float* gprow = gp + (size_t)(b * IC + ct * 16 + lo) * M;

    v8f acc = {0.f, 0.f, 0.f, 0.f, 0.f, 0.f, 0.f, 0.f};
    for (int m0 = 0; m0 < M; m0 += 4) {
        int mm = m0 + 2 * hi;
        v2f a;
        a.x = fmaxf(stl + sps[mm], 0.f) * invM;
        a.y = fmaxf(stl + sps[mm + 1], 0.f) * invM;
        v2f bf = *(const v2f*)&gprow[mm];
        acc = WMMA_F32X4(a, bf, acc);
    }

    float* yp = Y + (size_t)(b * IC + ct * 16 + lo) * N + nt * 16 + 8 * hi;
    *(float4*)yp       = make_float4(acc[0], acc[1], acc[2], acc[3]);
    *(float4*)(yp + 4) = make_float4(acc[4], acc[5], acc[6], acc[7]);
}

// w_y[b,co,n] = W_b[co] + sum_ic W_w[co,ic] * Y[b,ic,n]   (WMMA f32)
__global__ void __launch_bounds__(256) k_wconv(const float* __restrict__ Y,
                                               const float* __restrict__ W_w,
                                               const float* __restrict__ W_b,
                                               float* __restrict__ out) {
    int wave = (blockIdx.x * blockDim.x + threadIdx.x) >> 5;  // 0..32767
    int lane = threadIdx.x & 31;
    int lo = lane & 15, hi = lane >> 4;
    int b   = wave >> 12;
    int cot = (wave >> 8) & 15;
    int nt  = wave & 255;

    v8f acc;
#pragma unroll
    for (int v = 0; v < 8; ++v) acc[v] = W_b[cot * 16 + 8 * hi + v];

    const float* yb = Y + (size_t)b * IC * N + nt * 16 + lo;
    for (int k = 0; k < IC; k += 4) {
        int kk = k + 2 * hi;
        v2f a = *(const v2f*)&W_w[(cot * 16 + lo) * IC + kk];
        v2f bf;
        bf.x = yb[(size_t)kk * N];
        bf.y = yb[(size_t)(kk + 1) * N];
        acc = WMMA_F32X4(a, bf, acc);
    }
#pragma unroll
    for (int v = 0; v < 8; ++v)
        out[(size_t)(b * C + cot * 16 + 8 * hi + v) * N + nt * 16 + lo] = acc[v];
}

// per-channel mean / rstd over (B,H,W)
__global__ void k_stats(const float* __restrict__ wy, float* __restrict__ mean,
                        float* __restrict__ rstd) {
    int co = blockIdx.x;  // 256 blocks
    float s = 0.f, sq = 0.f;
    for (int i = threadIdx.x; i < B * N; i += 256) {
        int b = i >> 12;
        int n = i & (N - 1);
        float v = wy[(size_t)(b * C + co) * N + n];
        s += v;
        sq += v * v;
    }
    __shared__ float sh0[256], sh1[256];
    sh0[threadIdx.x] = s;
    sh1[threadIdx.x] = sq;
    __syncthreads();
    for (int off = 128; off > 0; off >>= 1) {
        if (threadIdx.x < off) {
            sh0[threadIdx.x] += sh0[threadIdx.x + off];
            sh1[threadIdx.x] += sh1[threadIdx.x + off];
        }
        __syncthreads();
    }
    if (threadIdx.x == 0) {
        float mn  = sh0[0] / (float)(B * N);
        float var = sh1[0] / (float)(B * N) - mn * mn;
        mean[co] = mn;
        rstd[co] = rsqrtf(var + BN_EPS);
    }
}

// normalize + affine + residual, in place on d_out
__global__ void k_bn(float* __restrict__ out, const float* __restrict__ x,
                     const float* __restrict__ mean, const float* __restrict__ rstd,
                     const float* __restrict__ gamma, const float* __restrict__ beta) {
    size_t i = (size_t)blockIdx.x * blockDim.x + threadIdx.x;
    int c = (int)((i >> 12) & (C - 1));
    float v = out[i];
    out[i] = (v - mean[c]) * rstd[c] * gamma[c] + beta[c] + x[i];
}

// ---------------------------------------------------------------------------
extern "C" void kernel_launch(void* const* d_in, const int* in_sizes, int n_in,
                              void* d_out, int out_size, void* d_ws, size_t ws_size,
                              hipStream_t stream) {
    (void)in_sizes; (void)n_in; (void)out_size; (void)ws_size;
    const float* x     = (const float*)d_in[0];
    const float* g_w   = (const float*)d_in[1];
    const float* g_b   = (const float*)d_in[2];
    const float* th_w  = (const float*)d_in[3];
    const float* th_b  = (const float*)d_in[4];
    const float* ph_w  = (const float*)d_in[5];
    const float* ph_b  = (const float*)d_in[6];
    const float* cp_w  = (const float*)d_in[7];
    const float* W_w   = (const float*)d_in[8];
    const float* W_b   = (const float*)d_in[9];
    const float* gamma = (const float*)d_in[10];
    const float* beta  = (const float*)d_in[11];

    float* ws  = (float*)d_ws;
    float* out = (float*)d_out;

    float* gp   = ws + GP_OFF;
    float* pp   = ws + PP_OFF;
    float* st   = ws + ST_OFF;
    float* sp   = ws + SP_OFF;
    float* tw   = ws + TW_OFF;
    float* tb   = ws + TB_OFF;
    float* mean = ws + MEAN_OFF;
    float* rstd = ws + RSTD_OFF;
    float* Y    = ws + Y_OFF;

    hipLaunchKernelGGL(k_redw, dim3(1), dim3(256), 0, stream, cp_w, th_w, th_b, tw, tb);
    hipLaunchKernelGGL(k_st, dim3(B * N / 256), dim3(256), 0, stream, x, tw, tb, st);
    hipLaunchKernelGGL(k_gphi, dim3(512), dim3(256), 0, stream, x, g_w, g_b, ph_w, ph_b, gp, pp);
    hipLaunchKernelGGL(k_sp, dim3(B * M / 256), dim3(256), 0, stream, pp, cp_w, sp);
    hipLaunchKernelGGL(k_attn, dim3(B * 256), dim3(256), 0, stream, gp, st, sp, Y);
    hipLaunchKernelGGL(k_wconv, dim3(4096), dim3(256), 0, stream, Y, W_w, W_b, out);
    hipLaunchKernelGGL(k_stats, dim3(C), dim3(256), 0, stream, out, mean, rstd);
    hipLaunchKernelGGL(k_bn, dim3(B * C * N / 256), dim3(256), 0, stream, out, x, mean, rstd,
                       gamma, beta);
}